// SageConvEncoder_90555090469222
// MI455X (gfx1250) — compile-verified
//
#include <hip/hip_runtime.h>

// GraphSAGE 3-layer encoder for MI455X (gfx1250, wave32).
// Memory-bound by edge scatter (~5 GB total -> ~0.22 ms @ 23.3 TB/s).
// GEMMs use exact-fp32 V_WMMA_F32_16X16X4_F32 (matrix pipe, no precision loss).
// Scatter-add uses inline-asm global_atomic_add_f32 (guaranteed native FP atomic,
// no CAS-loop fallback).

#define DFEAT 128

typedef __attribute__((ext_vector_type(2))) float v2f;
typedef __attribute__((ext_vector_type(8))) float v8f;

// ---------------- degree count ----------------
__global__ __launch_bounds__(256) void sage_count(const int* __restrict__ dst,
                                                  unsigned* __restrict__ cnt,
                                                  int n_edges) {
  int e = blockIdx.x * 256 + threadIdx.x;
  if (e < n_edges) atomicAdd(&cnt[dst[e]], 1u);
}

__global__ __launch_bounds__(256) void sage_inv(const unsigned* __restrict__ cnt,
                                                float* __restrict__ inv, int n) {
  int i = blockIdx.x * 256 + threadIdx.x;
  if (i < n) {
    unsigned c = cnt[i];
    inv[i] = 1.0f / (float)(c ? c : 1u);
  }
}

// ---------------- edge scatter-add ----------------
// one lane per (edge, float4 chunk): the 32 lanes of a wave cover one edge's
// 128-dim row -> uniform src/dst loads, 512B-coalesced gather + atomics.
__global__ __launch_bounds__(256) void sage_scatter(const float* __restrict__ xin,
                                                    const int* __restrict__ src,
                                                    const int* __restrict__ dst,
                                                    float* __restrict__ msg,
                                                    int n_edges) {
  unsigned gid = blockIdx.x * 256u + threadIdx.x;
  int e = (int)(gid >> 5);
  int c = (int)((gid & 31u) << 2);
  if (e >= n_edges) return;
  int s = src[e];
  int d = dst[e];
  const float4 v = *(const float4*)(xin + (size_t)s * DFEAT + c);
  // byte offset into msg; max ~51.2 MB, fits the signed 32-bit VGPR offset of
  // GVS addressing (SGPR64 base + VGPR32 offset).
  unsigned voff = (unsigned)(((size_t)d * DFEAT + c) * sizeof(float));
  // Four no-return global_atomic_add_f32 (STOREcnt-tracked; S_ENDPGM implies
  // wait-idle, so no explicit s_wait_storecnt needed).
  asm volatile(
      "global_atomic_add_f32 %0, %1, %5\n\t"
      "global_atomic_add_f32 %0, %2, %5 offset:4\n\t"
      "global_atomic_add_f32 %0, %3, %5 offset:8\n\t"
      "global_atomic_add_f32 %0, %4, %5 offset:12"
      :
      : "v"(voff), "v"(v.x), "v"(v.y), "v"(v.z), "v"(v.w), "s"(msg)
      : "memory");
}

// ---------------- fused SAGE layer GEMM ----------------
// out[16 rows] = mean @ Wl + x @ Wr + bias (optional ReLU)
// block = 256 threads = 8 wave32; wave w computes cols [16w, 16w+16).
// A fragments (16x4 f32): lanes 0-15 -> (M=lane, K=k0+{0,1}); lanes 16-31 -> (M=lane-16, K=k0+{2,3}).
// B fragments (4x16 f32): lanes 0-15 -> (K=k0+{0,1}, N=lane); lanes 16-31 -> (K=k0+{2,3}, N=lane-16).
__global__ __launch_bounds__(256) void sage_gemm(
    const float* __restrict__ msg, const float* __restrict__ inv,
    const float* __restrict__ xin, const float* __restrict__ Wl,
    const float* __restrict__ Wr, const float* __restrict__ bias,
    float* __restrict__ out, int relu) {
  __shared__ float sM[16][DFEAT + 4];  // mean rows (pad 4 dwords: conflict-free)
  __shared__ float sX[16][DFEAT + 4];  // x rows

  const int tid = threadIdx.x;
  const size_t rowBase = (size_t)blockIdx.x * 16;

  // cooperative stage: 16 rows x 128 cols, float4 granularity (512 float4 / 256 thr)
  for (int i = tid; i < 16 * (DFEAT / 4); i += 256) {
    int r = i >> 5;            // 32 float4 per row
    int c = (i & 31) << 2;
    float s = inv[rowBase + r];
    float4 m = *(const float4*)(msg + (rowBase + r) * DFEAT + c);
    float4 x = *(const float4*)(xin + (rowBase + r) * DFEAT + c);
    sM[r][c + 0] = m.x * s; sM[r][c + 1] = m.y * s;
    sM[r][c + 2] = m.z * s; sM[r][c + 3] = m.w * s;
    sX[r][c + 0] = x.x; sX[r][c + 1] = x.y;
    sX[r][c + 2] = x.z; sX[r][c + 3] = x.w;
  }
  __syncthreads();

  const int lane = tid & 31;
  const int half = lane >> 4;           // 0: K pair {0,1}; 1: K pair {2,3}
  const int mrow = lane & 15;           // A row / D col within tile
  const int col = (tid >> 5) * 16 + mrow;  // global output column for this lane

  v8f acc;
  {
    float bv = bias[col];
#pragma unroll
    for (int r = 0; r < 8; ++r) acc[r] = bv;
  }

#pragma unroll 8
  for (int k0 = 0; k0 < DFEAT; k0 += 4) {
    const int ka = k0 + 2 * half;
    v2f aM, aX, bL, bR;
    aM[0] = sM[mrow][ka];  aM[1] = sM[mrow][ka + 1];
    aX[0] = sX[mrow][ka];  aX[1] = sX[mrow][ka + 1];
    bL[0] = Wl[(size_t)ka * DFEAT + col];
    bL[1] = Wl[(size_t)(ka + 1) * DFEAT + col];
    bR[0] = Wr[(size_t)ka * DFEAT + col];
    bR[1] = Wr[(size_t)(ka + 1) * DFEAT + col];
    acc = __builtin_amdgcn_wmma_f32_16x16x4_f32(false, aM, false, bL,
                                                (short)0, acc, false, false);
    acc = __builtin_amdgcn_wmma_f32_16x16x4_f32(false, aX, false, bR,
                                                (short)0, acc, false, false);
  }

  // C/D layout: VGPR r -> row (r + 8*half), lanes mod 16 -> col
#pragma unroll
  for (int r = 0; r < 8; ++r) {
    float v = acc[r];
    if (relu) v = fmaxf(v, 0.0f);
    out[(rowBase + r + 8 * half) * DFEAT + col] = v;
  }
}

extern "C" void kernel_launch(void* const* d_in, const int* in_sizes, int n_in,
                              void* d_out, int out_size, void* d_ws, size_t ws_size,
                              hipStream_t stream) {
  const float* x     = (const float*)d_in[0];
  const int*   ei    = (const int*)d_in[1];   // [2, E] int (harness: integer -> int*)
  const float* Wagg  = (const float*)d_in[2]; // [3, 128, 128]
  const float* Wroot = (const float*)d_in[3]; // [3, 128, 128]
  const float* b     = (const float*)d_in[4]; // [3, 128]

  const int N = in_sizes[0] / DFEAT;   // 100000 (divisible by 16)
  const int E = in_sizes[1] / 2;       // 1600000
  const int* src = ei;
  const int* dst = ei + E;

  float* out = (float*)d_out;          // also serves as hidden-state buffer (row-local updates)
  float* msg = (float*)d_ws;                         // N * 128 floats
  float* invc = msg + (size_t)N * DFEAT;             // N floats
  unsigned* cnt = (unsigned*)(invc + N);             // N u32

  // degree -> reciprocal (dst is layer-invariant: compute once)
  hipMemsetAsync(cnt, 0, (size_t)N * sizeof(unsigned), stream);
  sage_count<<<(E + 255) / 256, 256, 0, stream>>>(dst, cnt, E);
  sage_inv<<<(N + 255) / 256, 256, 0, stream>>>(cnt, invc, N);

  const int rowTiles = N / 16;
  const int scatterBlocks = (E * 32 + 255) / 256;

  for (int layer = 0; layer < 3; ++layer) {
    const float* in = (layer == 0) ? x : out;
    hipMemsetAsync(msg, 0, (size_t)N * DFEAT * sizeof(float), stream);
    sage_scatter<<<scatterBlocks, 256, 0, stream>>>(in, src, dst, msg, E);
    sage_gemm<<<rowTiles, 256, 0, stream>>>(
        msg, invc, in,
        Wagg + (size_t)layer * DFEAT * DFEAT,
        Wroot + (size_t)layer * DFEAT * DFEAT,
        b + (size_t)layer * DFEAT,
        out, (layer < 2) ? 1 : 0);
  }
}